// STTEncoder_66803921322694
// MI455X (gfx1250) — compile-verified
//
#include <hip/hip_runtime.h>
#include <stdint.h>

// ---------------------------------------------------------------------------
// Multi-scale foveation token extractor for MI455X (gfx1250).
// out[b][t][c][m][n] = mean over s x s box of img[b][c] at (y0+m*s, x0+n*s).
//
// Data path (CDNA5):
//   global -> LDS :  TENSOR_LOAD_TO_LDS (TDM, D# built per ISA 8.3/8.4),
//                    double-buffered per wave, S_WAIT_TENSORCNT pipelined.
//   LDS -> VGPR   :  ds_load_b64/b128, column pooling with VALU pk_add.
//   reduction     :  V_WMMA_F32_16X16X4_F32 accumulate (0/1 selector A).
// ---------------------------------------------------------------------------

typedef __attribute__((ext_vector_type(2))) float    v2f;
typedef __attribute__((ext_vector_type(8))) float    v8f;
typedef __attribute__((ext_vector_type(4))) uint32_t u32x4;
typedef __attribute__((ext_vector_type(4))) int      i32x4;
typedef __attribute__((ext_vector_type(8))) int      i32x8;

#define PAT  1280
#define NTOK 172

// Replicates _build_token_pattern(): corner (x,y) and stride s of token t.
// Uniform per block (blockIdx-derived) -> pure SALU, ~200 iterations.
__device__ __forceinline__ void token_info(int t, int& x, int& y, int& s) {
    const int STR[5] = {1, 2, 4, 6, 8};
    const int GRD[5] = {4, 4, 6, 8, 10};
    int n = 0, plo = 0, phi = 0;
    for (int ks = 0; ks < 5; ++ks) {
        const int ss = STR[ks], g = GRD[ks];
        const int cov = g * 16 * ss;
        const int off = (PAT - cov) / 2;
        for (int j = 0; j < g; ++j) {
            for (int i = 0; i < g; ++i) {
                const int xx = off + i * 16 * ss;
                const int yy = off + j * 16 * ss;
                if (ks > 0 && xx >= plo && xx + 16 * ss <= phi &&
                    yy >= plo && yy + 16 * ss <= phi)
                    continue;                      // covered by finer scale
                if (n == t) { x = xx; y = yy; s = ss; return; }
                ++n;
            }
        }
        plo = off; phi = off + cov;
    }
    x = 0; y = 0; s = 1; // unreachable
}

// ---------------------------------------------------------------------------
// TDM: DMA one 16-row x (16*S)-col f32 tile chunk from global into LDS.
// D# bit packing per CDNA5 ISA 8.3 (group 0) / 8.4 (group 1):
//   g0: [1:0]=count=1, [63:32]=lds_addr, [120:64]=global byte addr, [127:126]=2
//   g1: data_size=2 (4B), tensor_dim0=16S, tensor_dim1=16,
//       tile_dim0=16S, tile_dim1=16, tensor_dim0_stride=1280.
// Groups 2/3 zero (2D tensor). Uniform operands -> SGPR groups.
// ---------------------------------------------------------------------------
template <int S>
__device__ __forceinline__ void tdm_load_chunk(float* lds_dst,
                                               const float* gsrc) {
    const uint32_t RW = 16u * S;                       // row width (elements)
    const uint64_t ga = (uint64_t)(uintptr_t)gsrc;     // 57-bit byte address
    const uint32_t la = (uint32_t)(uintptr_t)lds_dst;  // low 32b = LDS offset

    u32x4 g0;
    g0.x = 1u;                                   // count=1, user descriptor
    g0.y = la;                                   // lds_addr
    g0.z = (uint32_t)ga;                         // global_addr[31:0]
    g0.w = (uint32_t)(ga >> 32) | (2u << 30);    // global_addr[56:32] | type=2

    i32x8 g1;
    g1[0] = 0x00020000;                          // data_size=4B, mask/flags=0
    g1[1] = (int)(RW << 16);                     // tensor_dim0[15:0] @ bit48
    g1[2] = (int)((RW >> 16) | (16u << 16));     // td0[31:16] | td1[15:0]
    g1[3] = (int)(RW << 16);                     // td1[31:16]=0 | tile_dim0
    g1[4] = 16;                                  // tile_dim1=16, tile_dim2=0
    g1[5] = PAT;                                 // tensor_dim0_stride[31:0]
    g1[6] = 0;                                   // stride0 hi | stride1 lo
    g1[7] = 0;                                   // stride1 hi

    const i32x4 z4 = {0, 0, 0, 0};
#if __clang_major__ >= 23
    const i32x8 z8 = {0, 0, 0, 0, 0, 0, 0, 0};
    __builtin_amdgcn_tensor_load_to_lds(g0, g1, z4, z4, z8, 0);
#else
    __builtin_amdgcn_tensor_load_to_lds(g0, g1, z4, z4, 0);
#endif
}

// Sum of S consecutive floats in LDS (alignment: row offsets are multiples of
// 64*S bytes, lane offsets of 4*S -> >=8B for S in {2,6}, 16B for {4,8}).
template <int S>
__device__ __forceinline__ float rowsum(const float* p) {
    if (S == 1) {
        return p[0];
    } else if (S == 2) {
        float2 v = *(const float2*)p;
        return v.x + v.y;
    } else if (S == 4) {
        float4 v = *(const float4*)p;
        return (v.x + v.y) + (v.z + v.w);
    } else if (S == 6) {
        float2 a = *(const float2*)p;
        float2 b = *(const float2*)(p + 2);
        float2 c = *(const float2*)(p + 4);
        return ((a.x + a.y) + (b.x + b.y)) + (c.x + c.y);
    } else { // S == 8
        float4 a = *(const float4*)p;
        float4 b = *(const float4*)(p + 4);
        return ((a.x + a.y) + (a.z + a.w)) + ((b.x + b.y) + (b.z + b.w));
    }
}

// One wave handles one (batch, token, channel): 16S x 16S tile -> 16x16.
// Double-buffered TDM chunks of 16 rows; K consumed 4 rows/WMMA.
//   A (16x4 selector): VGPR0 = K{0,2}, VGPR1 = K{1,3}; M = lane&15.
//   B (4x16 pooled):   VGPR0 = K{0,2}, VGPR1 = K{1,3}; N = lane&15.
//   C/D: VGPR v -> row M = v + 8*(lane>=16), col N = lane&15.
template <int S>
__device__ __forceinline__ void process_token(const float* __restrict__ base,
                                              float* __restrict__ o,
                                              int lane, float* chanbuf) {
    const int mn   = lane & 15;
    const int rsel = (lane & 16) ? 2 : 0;
    const int RW   = 16 * S;                       // floats per tile row

    tdm_load_chunk<S>(chanbuf, base);              // prime chunk 0

    v8f acc = {};
    for (int chunk = 0; chunk < S; ++chunk) {
        if (chunk + 1 < S)                         // issue next chunk
            tdm_load_chunk<S>(chanbuf + ((chunk + 1) & 1) * 2048,
                              base + (size_t)(chunk + 1) * 16 * PAT);
        // TDM per-wave in-order: <=1 outstanding => oldest (chunk) landed.
        if (chunk + 1 < S) __builtin_amdgcn_s_wait_tensorcnt((short)1);
        else               __builtin_amdgcn_s_wait_tensorcnt((short)0);
        asm volatile("" ::: "memory");             // keep ds_loads after wait

        const float* buf = chanbuf + (chunk & 1) * 2048;
#pragma unroll
        for (int kk = 0; kk < 4; ++kk) {
            const int krow = kk * 4 + rsel;
            const float* rp = buf + krow * RW + mn * S;
            v2f bv;
            bv.x = rowsum<S>(rp);                  // pooled row krow
            bv.y = rowsum<S>(rp + RW);             // pooled row krow+1

            const int arow = chunk * 16 + krow;    // absolute image row
            v2f av;                                // A[m][kk] = (row/S == m)
            av.x = ((arow / S)       == mn) ? 1.0f : 0.0f;
            av.y = (((arow + 1) / S) == mn) ? 1.0f : 0.0f;

            acc = __builtin_amdgcn_wmma_f32_16x16x4_f32(
                /*neg_a=*/false, av, /*neg_b=*/false, bv,
                /*c_mod=*/(short)0, acc, /*reuse_a=*/false, /*reuse_b=*/false);
        }
        // DS reads of this buffer must finish before TDM reuses it next iter
        // (TDM LDS writes are unordered vs this wave's ds_loads).
        asm volatile("s_wait_dscnt 0x0" ::: "memory");
    }

    const float inv_area = 1.0f / (float)(S * S);
    const int mbase = (lane & 16) ? 8 : 0;
#pragma unroll
    for (int v = 0; v < 8; ++v)
        o[(mbase + v) * 16 + mn] = acc[v] * inv_area;
}

extern "C" __global__ void __launch_bounds__(96)
fovea_wmma_kernel(const float* __restrict__ img, float* __restrict__ out) {
    extern __shared__ float smem[];               // 3 ch x 2 bufs x 2048 f32
    const int bt   = blockIdx.x;
    const int b    = bt / NTOK;
    const int t    = bt - b * NTOK;
    const int c    = threadIdx.x >> 5;            // wave id == channel
    const int lane = threadIdx.x & 31;

    int x0, y0, s;
    token_info(t, x0, y0, s);

    const float* base = img + (((size_t)b * 3 + c) * PAT + y0) * PAT + x0;
    float*       o    = out + (((size_t)b * NTOK + t) * 3 + c) * 256;
    float* chanbuf    = smem + c * 4096;          // 2 x 2048-float buffers

    switch (s) {                                  // uniform across block
        case 1: process_token<1>(base, o, lane, chanbuf); break;
        case 2: process_token<2>(base, o, lane, chanbuf); break;
        case 4: process_token<4>(base, o, lane, chanbuf); break;
        case 6: process_token<6>(base, o, lane, chanbuf); break;
        default: process_token<8>(base, o, lane, chanbuf); break;
    }
}

extern "C" void kernel_launch(void* const* d_in, const int* in_sizes, int n_in,
                              void* d_out, int out_size, void* d_ws, size_t ws_size,
                              hipStream_t stream) {
    (void)n_in; (void)out_size; (void)d_ws; (void)ws_size;
    const float* img = (const float*)d_in[0];
    float*       out = (float*)d_out;
    const int nB = in_sizes[0] / (3 * PAT * PAT);   // batch count (16)
    dim3 grid(nB * NTOK);                           // one block per (b, token)
    dim3 block(96);                                 // 3 waves = 3 channels
    const size_t shmem = 3 * 2 * 2048 * sizeof(float);  // 48 KB
    fovea_wmma_kernel<<<grid, block, shmem, stream>>>(img, out);
}